// SparsemaxBisectLoss_88476326298301
// MI455X (gfx1250) — compile-verified
//
#include <hip/hip_runtime.h>
#include <hip/hip_bf16.h>
#include <stdint.h>

// ---------------------------------------------------------------------------
// Sparsemax-bisect loss, MI455X (gfx1250, wave32).
//   - one HBM pass over X (524 MB -> ~22.5 us roofline @ 23.3 TB/s)
//   - row staged in LDS via CDNA5 async global->LDS DMA (ASYNCcnt),
//     double-buffered in 2 x 128 KB of the 320 KB WGP LDS; next-row DMA
//     issued right after the current row becomes visible, so it overlaps
//     the max/compaction/bisection compute
//   - candidate compaction: only x > max-1 can contribute to any relu-sum,
//     so the 50 bisection iterations run on ~tens of values (wave0 only,
//     shfl_xor reductions, no barriers in the loop)
// ---------------------------------------------------------------------------

#define N_ROWS 4096
#define NC     32000
#define NITERS 50
#define TPB    256
#define NWAVE  (TPB / 32)
#define NCHUNK (NC / 4)                    // 8000 float4 tiles per row
#define CPT    ((NCHUNK + TPB - 1) / TPB)  // 32 chunks per thread (padded)
#define CAP    1024                        // candidate buffer capacity
#define GRID   1024
#define SMEM_FLOATS (2 * NC + CAP + 32)

#if defined(__HIP_DEVICE_COMPILE__) && __has_builtin(__builtin_amdgcn_global_load_async_to_lds_b128)
#define HAVE_ASYNC_LDS 1
#else
#define HAVE_ASYNC_LDS 0
#endif

// Exact pointee type the builtin expects: 16-byte int vector,
// global (AS1) source / LDS (AS3) destination.
typedef int i4v __attribute__((__vector_size__(16)));
typedef __attribute__((address_space(1))) i4v* g_i4p;
typedef __attribute__((address_space(3))) i4v* l_i4p;

__device__ __forceinline__ void wait_async0() {
#if HAVE_ASYNC_LDS
#if __has_builtin(__builtin_amdgcn_s_wait_asynccnt)
  __builtin_amdgcn_s_wait_asynccnt(0);
#else
  asm volatile("s_wait_asynccnt 0x0" ::: "memory");
#endif
#endif
}

// Stage one 128 KB row (8000 x b128) into LDS, coalesced across the block.
__device__ __forceinline__ void copy_row_to_lds(const float* __restrict__ g,
                                                float* lbuf, int t) {
#if HAVE_ASYNC_LDS
#pragma unroll
  for (int k = 0; k < CPT; ++k) {
    int c = t + k * TPB;
    if (c >= NCHUNK) c = NCHUNK - 1;  // uniform issue count (idempotent dup)
    __builtin_amdgcn_global_load_async_to_lds_b128(
        (g_i4p)(uintptr_t)(g + 4 * c),
        (l_i4p)(uint32_t)(uintptr_t)(lbuf + 4 * c),
        0, 0);
  }
#else
#pragma unroll
  for (int k = 0; k < CPT; ++k) {
    int c = t + k * TPB;
    if (c < NCHUNK)
      *(float4*)(lbuf + 4 * c) = *(const float4*)(g + 4 * c);
  }
#endif
}

__global__ void __launch_bounds__(TPB, 1)
sparsemax_rows_kernel(const float* __restrict__ X,
                      const int* __restrict__ target,
                      float* __restrict__ row_loss) {
  extern __shared__ float smem[];
  float* bufA = smem;                       // NC floats
  float* bufB = smem + NC;                  // NC floats
  float* cand = smem + 2 * NC;              // CAP floats
  float* redf = smem + 2 * NC + CAP;        // NWAVE floats (max partials)
  int*   wcnt = (int*)(smem + 2 * NC + CAP + 8);   // wave candidate totals
  int*   woff = (int*)(smem + 2 * NC + CAP + 16);  // wave exclusive offsets
  float* ctrl = smem + 2 * NC + CAP + 24;   // [0] = row max
  int*   ictl = (int*)(smem + 2 * NC + CAP + 25);  // [0] = ncand

  const int t    = threadIdx.x;
  const int lane = t & 31;
  const int wid  = t >> 5;
  const int stride = gridDim.x;

  int row = blockIdx.x;
  if (row >= N_ROWS) return;

  int p = 0;
  copy_row_to_lds(X + (size_t)row * NC, bufA, t);

  for (; row < N_ROWS; row += stride) {
    float* lbuf = p ? bufB : bufA;
    float* nbuf = p ? bufA : bufB;

    // wait for this row's DMA (per-wave ASYNCcnt), make LDS visible to all
    wait_async0();
    __syncthreads();

    // prefetch next row into the free buffer; overlaps all compute below
    const int nrow = row + stride;
    if (nrow < N_ROWS) copy_row_to_lds(X + (size_t)nrow * NC, nbuf, t);

    const float4* lb4 = (const float4*)lbuf;

    // ---- pass 1: row max -------------------------------------------------
    float m = -3.402823466e+38f;
#pragma unroll
    for (int k = 0; k < CPT; ++k) {
      int c = t + k * TPB;
      if (c < NCHUNK) {
        float4 v = lb4[c];
        m = fmaxf(m, fmaxf(fmaxf(v.x, v.y), fmaxf(v.z, v.w)));
      }
    }
    for (int s = 16; s; s >>= 1) m = fmaxf(m, __shfl_xor(m, s, 32));
    if (lane == 0) redf[wid] = m;
    __syncthreads();
    if (t == 0) {
      float mm = redf[0];
      for (int w = 1; w < NWAVE; ++w) mm = fmaxf(mm, redf[w]);
      ctrl[0] = mm;
    }
    __syncthreads();
    const float maxv = ctrl[0];
    const float thr  = maxv - 1.0f;  // = tau_lo; only x > thr ever matter

    // ---- pass 2: count candidates, deterministic prefix-scan compaction --
    int cnt = 0;
#pragma unroll
    for (int k = 0; k < CPT; ++k) {
      int c = t + k * TPB;
      if (c < NCHUNK) {
        float4 v = lb4[c];
        cnt += (v.x > thr) + (v.y > thr) + (v.z > thr) + (v.w > thr);
      }
    }
    int incl = cnt;
    for (int d = 1; d < 32; d <<= 1) {
      int y = __shfl_up(incl, d, 32);
      if (lane >= d) incl += y;
    }
    if (lane == 31) wcnt[wid] = incl;
    __syncthreads();
    if (t == 0) {
      int run = 0;
      for (int w = 0; w < NWAVE; ++w) { woff[w] = run; run += wcnt[w]; }
      ictl[0] = run;
    }
    __syncthreads();
    const int ncand = ictl[0];

    if (ncand <= CAP) {
      int off = woff[wid] + (incl - cnt);
#pragma unroll
      for (int k = 0; k < CPT; ++k) {
        int c = t + k * TPB;
        if (c < NCHUNK) {
          float4 v = lb4[c];
          if (v.x > thr) cand[off++] = v.x;
          if (v.y > thr) cand[off++] = v.y;
          if (v.z > thr) cand[off++] = v.z;
          if (v.w > thr) cand[off++] = v.w;
        }
      }
    }
    __syncthreads();

    // ---- bisection on candidates: wave 0 only, barrier-free --------------
    if (wid == 0) {
      const float* src = (ncand <= CAP) ? cand : lbuf;  // fallback: full row
      const int    n   = (ncand <= CAP) ? ncand : NC;

      const float tau_lo0 = maxv - 1.0f;
      const float tau_hi  = maxv - (1.0f / (float)NC);

      float s = 0.0f;
      for (int i = lane; i < n; i += 32) s += fmaxf(src[i] - tau_lo0, 0.0f);
      for (int sh = 16; sh; sh >>= 1) s += __shfl_xor(s, sh, 32);
      const float f_lo = s - 1.0f;

      float tau_lo = tau_lo0;
      float dm     = tau_hi - tau_lo0;
      float tau_m  = tau_lo0;
      for (int it = 0; it < NITERS; ++it) {
        dm *= 0.5f;
        tau_m = tau_lo + dm;
        float fm = 0.0f;
        for (int i = lane; i < n; i += 32) fm += fmaxf(src[i] - tau_m, 0.0f);
        for (int sh = 16; sh; sh >>= 1) fm += __shfl_xor(fm, sh, 32);
        fm -= 1.0f;
        if (fm * f_lo >= 0.0f) tau_lo = tau_m;
      }

      // loss = 0.5*(1 - sum p^2) + sum(p*x) - x_tgt,  p = r/S, r = relu(x-tau_m)
      float S = 0.0f, S2 = 0.0f, SX = 0.0f;
      for (int i = lane; i < n; i += 32) {
        float x = src[i];
        float r = fmaxf(x - tau_m, 0.0f);
        S += r; S2 += r * r; SX += r * x;
      }
      for (int sh = 16; sh; sh >>= 1) {
        S  += __shfl_xor(S,  sh, 32);
        S2 += __shfl_xor(S2, sh, 32);
        SX += __shfl_xor(SX, sh, 32);
      }
      if (lane == 0) {
        const int   tgt = target[row];
        const float xt  = lbuf[tgt];
        const float inv = 1.0f / S;
        row_loss[row] = 0.5f * (1.0f - S2 * inv * inv) + SX * inv - xt;
      }
    }
    __syncthreads();
    p ^= 1;
  }
}

__global__ void mean_reduce_kernel(const float* __restrict__ ws,
                                   float* __restrict__ out) {
  __shared__ float r[NWAVE];
  const int t = threadIdx.x, lane = t & 31, wid = t >> 5;
  float s = 0.0f;
  for (int i = t; i < N_ROWS; i += TPB) s += ws[i];
  for (int sh = 16; sh; sh >>= 1) s += __shfl_xor(s, sh, 32);
  if (lane == 0) r[wid] = s;
  __syncthreads();
  if (t == 0) {
    float tot = 0.0f;
    for (int w = 0; w < NWAVE; ++w) tot += r[w];
    out[0] = tot * (1.0f / (float)N_ROWS);
  }
}

extern "C" void kernel_launch(void* const* d_in, const int* in_sizes, int n_in,
                              void* d_out, int out_size, void* d_ws, size_t ws_size,
                              hipStream_t stream) {
  (void)in_sizes; (void)n_in; (void)out_size; (void)ws_size;
  const float* X      = (const float*)d_in[0];
  const int*   target = (const int*)d_in[1];
  float*       losses = (float*)d_ws;          // 4096 floats of scratch
  float*       out    = (float*)d_out;

  const size_t shbytes = (size_t)SMEM_FLOATS * sizeof(float);  // ~260 KB < 320 KB WGP LDS
  sparsemax_rows_kernel<<<GRID, TPB, shbytes, stream>>>(X, target, losses);
  mean_reduce_kernel<<<1, TPB, 0, stream>>>(losses, out);
}